// MovingAverageClustering_76003741270541
// MI455X (gfx1250) — compile-verified
//
#include <hip/hip_runtime.h>

// MovingAverageClustering on MI455X (gfx1250, wave32, WMMA f32 16x16x4).
// N=100000 points, K=512 centroids, D=256, fp32.

#define NPTS   100000
#define KCENT  512
#define DDIM   256
#define DECAY  0.99f

// LDS x-tile row stride (floats). 268*4B is a multiple of 16 (float4 stores OK)
// and 268 % 64 == 12, so rows 0..15 land on 16 distinct banks for the b64
// A-fragment loads -> conflict-free.
#define XS_STRIDE 268

typedef float v2f __attribute__((ext_vector_type(2)));
typedef float v8f __attribute__((ext_vector_type(8)));

__device__ __forceinline__ unsigned orderable_f32(float s) {
  unsigned u = __float_as_uint(s);
  return (u & 0x80000000u) ? ~u : (u | 0x80000000u);
}

__device__ __forceinline__ unsigned long long ullmin2(unsigned long long a,
                                                      unsigned long long b) {
  return b < a ? b : a;
}

// ---------------------------------------------------------------------------
// K0: zero workspace accumulators, set match flag = 1, cnorm1[k] = |c_k|^2
// grid = K, block = D
// ---------------------------------------------------------------------------
__global__ __launch_bounds__(DDIM)
void mac_init_kernel(const float* __restrict__ cent,
                     float* __restrict__ cnorm1,
                     float* __restrict__ sums,
                     float* __restrict__ counts,
                     int*   __restrict__ flag) {
  const int k = blockIdx.x;
  const int t = threadIdx.x;
  const float c = cent[(size_t)k * DDIM + t];
  sums[(size_t)k * DDIM + t] = 0.0f;

  __shared__ float red[DDIM];
  red[t] = c * c;
  __syncthreads();
  for (int s = DDIM / 2; s > 0; s >>= 1) {
    if (t < s) red[t] += red[t + s];
    __syncthreads();
  }
  if (t == 0) {
    cnorm1[k] = red[0];
    counts[k] = 0.0f;
    if (k == 0) *flag = 1;
  }
}

// ---------------------------------------------------------------------------
// Assignment GEMM + argmin.  8 waves: 2 (M) x 4 (N); each wave owns TWO
// 16x16 WMMA tiles in N (one ds_load_b64 A-fragment feeds two v_wmma).
// Block covers 32 rows of x; kc loops over K in steps of 128 (4 iters).
// Per-lane running-best keys defer the cross-lane argmin reduction to a
// single post-loop pass (no ds_bpermute / ds_min_u64 in the hot loop).
// PASS1: fused segment-sum + prev-assignment match check.
// PASS2: writes final assignments (as float) to out.
// ---------------------------------------------------------------------------
template <bool PASS1>
__global__ __launch_bounds__(256)
void mac_assign_kernel(const float* __restrict__ x,
                       const float* __restrict__ cent,
                       const float* __restrict__ cnorm,
                       const int*   __restrict__ prev,
                       float* __restrict__ sums,
                       float* __restrict__ counts,
                       int*   __restrict__ flag,
                       float* __restrict__ out_assign) {
  __shared__ float Xs[32 * XS_STRIDE];
  __shared__ unsigned long long best[32];
  __shared__ int asg[32];

  const int tid   = threadIdx.x;
  const int lane  = tid & 31;
  const int wave  = tid >> 5;
  const int waveM = wave >> 2;   // 0..1
  const int waveN = wave & 3;    // 0..3
  const int rowBase = blockIdx.x * 32;   // N = 32 * 3125 exactly

  // ---- stage x tile: 32 rows x 256 cols, float4 coalesced ----
  {
    const float4* xg = (const float4*)(x + (size_t)rowBase * DDIM);
#pragma unroll
    for (int i = 0; i < 8; ++i) {
      const int idx = tid + i * 256;   // float4 index, 0..2047
      const int r   = idx >> 6;        // 64 float4 per row
      const int c4  = idx & 63;
      float4 v = xg[(size_t)r * 64 + c4];
      *(float4*)&Xs[r * XS_STRIDE + c4 * 4] = v;
    }
  }
  if (tid < 32) best[tid] = ~0ull;
  __syncthreads();

  // A fragment (16x4 f32): lanes 0-15 rows 0..15 k={d,d+1}; lanes 16-31 same
  // rows, k={d+2,d+3}.  B fragment (4x16) mirrors with lane%16 = column.
  const int aoff = (lane < 16) ? 0 : 2;
  const float* xrow = &Xs[(waveM * 16 + (lane & 15)) * XS_STRIDE];

  unsigned long long bk0[8], bk1[8];
#pragma unroll
  for (int i = 0; i < 8; ++i) { bk0[i] = ~0ull; bk1[i] = ~0ull; }

  for (int kc = 0; kc < KCENT; kc += 128) {
    const int ncol0 = kc + waveN * 32 + (lane & 15);
    const int ncol1 = ncol0 + 16;
    const float* crow0 = cent + (size_t)ncol0 * DDIM;
    const float* crow1 = cent + (size_t)ncol1 * DDIM;

    v8f acc0 = {0.f, 0.f, 0.f, 0.f, 0.f, 0.f, 0.f, 0.f};
    v8f acc1 = {0.f, 0.f, 0.f, 0.f, 0.f, 0.f, 0.f, 0.f};
#pragma unroll 8
    for (int d = 0; d < DDIM; d += 4) {
      v2f a  = *(const v2f*)(xrow  + d + aoff);   // ds_load_b64, bank-clean
      v2f b0 = *(const v2f*)(crow0 + d + aoff);   // global_load_b64, L2-hot
      v2f b1 = *(const v2f*)(crow1 + d + aoff);
      acc0 = __builtin_amdgcn_wmma_f32_16x16x4_f32(
          false, a, false, b0, (short)0, acc0, false, false);
      acc1 = __builtin_amdgcn_wmma_f32_16x16x4_f32(
          false, a, false, b1, (short)0, acc1, false, false);
    }

    // score = |c|^2 - 2 * <x, c>  (row-constant |x|^2 dropped for argmin).
    // Defer cross-lane reduction: keep per-lane running best (score, col)
    // keys; u64 min resolves ties to the lowest column, like jnp.argmin.
    const float cn0 = cnorm[ncol0];
    const float cn1 = cnorm[ncol1];
#pragma unroll
    for (int i = 0; i < 8; ++i) {
      const float s0 = cn0 - 2.0f * acc0[i];
      const float s1 = cn1 - 2.0f * acc1[i];
      bk0[i] = ullmin2(bk0[i], ((unsigned long long)orderable_f32(s0) << 32) |
                                   (unsigned)ncol0);
      bk1[i] = ullmin2(bk1[i], ((unsigned long long)orderable_f32(s1) << 32) |
                                   (unsigned)ncol1);
    }
  }

  // ---- one cross-lane argmin reduction per block ----
#pragma unroll
  for (int i = 0; i < 8; ++i) {
    unsigned long long key = ullmin2(bk0[i], bk1[i]);
    key = ullmin2(key, __shfl_xor(key, 1, 32));
    key = ullmin2(key, __shfl_xor(key, 2, 32));
    key = ullmin2(key, __shfl_xor(key, 4, 32));
    key = ullmin2(key, __shfl_xor(key, 8, 32));
    if ((lane & 15) == 0) {
      const int rloc = waveM * 16 + i + ((lane >= 16) ? 8 : 0);
      atomicMin(&best[rloc], key);   // ds_min_u64, cross-wave combine
    }
  }
  __syncthreads();

  if (PASS1) {
    if (tid < 32) {
      const int a = (int)(best[tid] & 0xffffffffull);
      asg[tid] = a;
      if (a != prev[rowBase + tid]) atomicAnd(flag, 0);
      atomicAdd(&counts[a], 1.0f);
    }
    __syncthreads();
    // fused segment-sum: rows already in LDS, one column per thread
    for (int r = 0; r < 32; ++r) {
      atomicAdd(&sums[(size_t)asg[r] * DDIM + tid], Xs[r * XS_STRIDE + tid]);
    }
  } else {
    if (tid < 32) {
      out_assign[rowBase + tid] = (float)(best[tid] & 0xffffffffull);
    }
  }
}

// ---------------------------------------------------------------------------
// K2: centroids_out = flag ? (1-DECAY)*c + DECAY*sums/counts : c
//     cnorm2[k] = |centroids_out[k]|^2 ; flag_out = flag
// grid = K, block = D
// ---------------------------------------------------------------------------
__global__ __launch_bounds__(DDIM)
void mac_update_kernel(const float* __restrict__ cent,
                       const float* __restrict__ sums,
                       const float* __restrict__ counts,
                       const int*   __restrict__ flag,
                       float* __restrict__ cent_out,
                       float* __restrict__ flag_out,
                       float* __restrict__ cnorm2) {
  const int k = blockIdx.x;
  const int t = threadIdx.x;
  const float c = cent[(size_t)k * DDIM + t];
  float v = c;
  if (*flag) {
    const float nc = sums[(size_t)k * DDIM + t] / counts[k];
    v = (1.0f - DECAY) * c + DECAY * nc;
  }
  cent_out[(size_t)k * DDIM + t] = v;

  __shared__ float red[DDIM];
  red[t] = v * v;
  __syncthreads();
  for (int s = DDIM / 2; s > 0; s >>= 1) {
    if (t < s) red[t] += red[t + s];
    __syncthreads();
  }
  if (t == 0) {
    cnorm2[k] = red[0];
    if (k == 0) flag_out[0] = (*flag) ? 1.0f : 0.0f;
  }
}

// ---------------------------------------------------------------------------
extern "C" void kernel_launch(void* const* d_in, const int* in_sizes, int n_in,
                              void* d_out, int out_size, void* d_ws, size_t ws_size,
                              hipStream_t stream) {
  (void)in_sizes; (void)n_in; (void)out_size; (void)ws_size;
  const float* x    = (const float*)d_in[0];
  const float* cent = (const float*)d_in[1];
  const int*   prev = (const int*)d_in[2];

  float* out = (float*)d_out;
  float* out_assign = out;                 // [0, N)
  float* out_flag   = out + NPTS;          // [N]
  float* out_cent   = out + NPTS + 1;      // [N+1, N+1+K*D)

  float* ws     = (float*)d_ws;
  float* cnorm1 = ws;                       // 512
  float* cnorm2 = ws + 512;                 // 512
  float* counts = ws + 1024;                // 512
  int*   flag   = (int*)(ws + 1536);        // 1 (padded)
  float* sums   = ws + 2048;                // K*D = 131072

  mac_init_kernel<<<KCENT, DDIM, 0, stream>>>(cent, cnorm1, sums, counts, flag);

  mac_assign_kernel<true><<<NPTS / 32, 256, 0, stream>>>(
      x, cent, cnorm1, prev, sums, counts, flag, nullptr);

  mac_update_kernel<<<KCENT, DDIM, 0, stream>>>(
      cent, sums, counts, flag, out_cent, out_flag, cnorm2);

  // Pass 2 against centroids_out: equals the updated-centroid argmin when
  // match==true, and reproduces pass-1 assignments when match==false.
  mac_assign_kernel<false><<<NPTS / 32, 256, 0, stream>>>(
      x, out_cent, cnorm2, nullptr, nullptr, nullptr, nullptr, out_assign);
}